// LocalSphericalNeuralOperator_6614249636092
// MI455X (gfx1250) — compile-verified
//
#include <hip/hip_runtime.h>
#include <math.h>

typedef __attribute__((ext_vector_type(2))) float v2f;
typedef __attribute__((ext_vector_type(8))) float v8f;

__device__ __forceinline__ v8f wmma_f32(v2f a, v2f b, v8f c) {
  return __builtin_amdgcn_wmma_f32_16x16x4_f32(false, a, false, b,
                                               (short)0, c, false, false);
}

__device__ __forceinline__ float gelu_tanh(float u) {
  return 0.5f * u * (1.f + tanhf(0.7978845608028654f * (u + 0.044715f * u * u * u)));
}

// ---------------------------------------------------------------------------
// wv[h][o][c*16+n] = sum_k w[o][c][k] * vals[k][h][n], zero-padded to Opad
// rows so the conv kernel's A-fragment loads are unconditional.
// ---------------------------------------------------------------------------
__global__ void build_wv_kernel(const float* __restrict__ w,    // [O][C][9]
                                const float* __restrict__ vals, // [9][Hout][16]
                                float* __restrict__ wv,         // [Hout][Opad][C*16]
                                int O, int Opad, int C, int Hout) {
  int idx = blockIdx.x * blockDim.x + threadIdx.x;
  int total = Hout * Opad * C * 16;
  if (idx >= total) return;
  int n = idx & 15;
  int t = idx >> 4;
  int c = t % C; t /= C;
  int o = t % Opad;
  int h = t / Opad;
  float s = 0.f;
  if (o < O) {
#pragma unroll
    for (int k = 0; k < 9; ++k)
      s += w[(o * C + c) * 9 + k] * vals[(k * Hout + h) * 16 + n];
  }
  wv[idx] = s;
}

// ---------------------------------------------------------------------------
// Per-latitude-row GEMM via V_WMMA_F32_16X16X4_F32, all dims compile-time.
// out[o][h][w] = sum_{c,n} wv[h][o][c*16+n] *
//                x[c][lat[h][n]][(off[h][n]+STRIDE*w) & (Win-1)]
// grid = (Wout/16, ceil(O/16), Hout), block = 32 (one wave per 16x16 D tile).
// Channel loop is 2-stage software pipelined: stage c+1 while 4 WMMAs of c run.
// ---------------------------------------------------------------------------
template <int O, int C, int Hin, int Win, int Hout, int Wout, int STRIDE>
__global__ __launch_bounds__(32) void disco_wmma_conv_kernel(
    const float* __restrict__ xin,  // [C][Hin][Win]
    const float* __restrict__ wv,   // [Hout][Opad][C*16]
    const int* __restrict__ lat,    // [Hout][16]
    const int* __restrict__ lon,    // [Hout][16]
    float* __restrict__ outp)       // [O][Hout][Wout]
{
  constexpr int OP = (O + 15) & ~15;
  constexpr int R = C * 16;
  constexpr int CHW = Hin * Win;

  const int wt = blockIdx.x;
  const int mt = blockIdx.y;
  const int h = blockIdx.z;
  const int lane = threadIdx.x & 31;
  const int half = lane >> 4;  // selects K pair {0,1} vs {2,3} inside each step
  const int l16 = lane & 15;
  const int m = mt * 16 + l16;  // A-frag row (M)
  const int n = wt * 16 + l16;  // B-frag col (N) == output longitude

  // Per-lane gather offsets for this lane's 8 neighbor taps (loop-invariant
  // over channels): kstep j in 0..3, element v in 0..1 -> nb = 4j + 2*half + v.
  const int* __restrict__ latrow = lat + h * 16;
  const int* __restrict__ lonrow = lon + h * 16;
  int off[8];
#pragma unroll
  for (int j = 0; j < 4; ++j) {
#pragma unroll
    for (int v = 0; v < 2; ++v) {
      const int nb = 4 * j + 2 * half + v;
      off[2 * j + v] = latrow[nb] * Win + ((lonrow[nb] + STRIDE * n) & (Win - 1));
    }
  }

  const float* __restrict__ arow = wv + ((h * OP + m) * R + half * 2);

  v8f acc = {};
  v2f a[4];
  float b[8];

  // Prologue: stage channel 0.
#pragma unroll
  for (int j = 0; j < 4; ++j) a[j] = *(const v2f*)(arow + 4 * j);
#pragma unroll
  for (int t = 0; t < 8; ++t) b[t] = xin[off[t]];

  for (int c = 0; c < C; ++c) {
    v2f ca[4];
    float cb[8];
#pragma unroll
    for (int j = 0; j < 4; ++j) ca[j] = a[j];
#pragma unroll
    for (int t = 0; t < 8; ++t) cb[t] = b[t];

    if (c + 1 < C) {  // stage next channel while WMMAs execute
      const float* an = arow + (c + 1) * 16;
      const float* xn = xin + (c + 1) * CHW;
#pragma unroll
      for (int j = 0; j < 4; ++j) a[j] = *(const v2f*)(an + 4 * j);
#pragma unroll
      for (int t = 0; t < 8; ++t) b[t] = xn[off[t]];
    }

#pragma unroll
    for (int j = 0; j < 4; ++j) {
      v2f bb;
      bb[0] = cb[2 * j];
      bb[1] = cb[2 * j + 1];
      acc = wmma_f32(ca[j], bb, acc);
    }
  }

#pragma unroll
  for (int r = 0; r < 8; ++r) {
    const int mm = mt * 16 + r + half * 8;
    if constexpr (O % 16 != 0) {
      if (mm < O) outp[(mm * Hout + h) * Wout + n] = acc[r];
    } else {
      outp[(mm * Hout + h) * Wout + n] = acc[r];
    }
  }
}

// ---------------------------------------------------------------------------
// Pointwise GEMM out[M][NP] = A[M][Kd] x B[Kd][NP] over NP = 128*256 pixels,
// fused GELU / residual selected at compile time. grid = (NP/16, M/16).
// ---------------------------------------------------------------------------
template <int M, int Kd, bool GELU, bool RES>
__global__ __launch_bounds__(32) void mlp_gemm_kernel(
    const float* __restrict__ A,      // [M][Kd]
    const float* __restrict__ B,      // [Kd][NP]
    float* __restrict__ outp,         // [M][NP]
    const float* __restrict__ resid)  // [M][NP] or nullptr
{
  constexpr int NP = 128 * 256;
  const int nt = blockIdx.x;
  const int mt = blockIdx.y;
  const int lane = threadIdx.x & 31;
  const int half = lane >> 4;
  const int l16 = lane & 15;
  const int n = nt * 16 + l16;
  const int m = mt * 16 + l16;

  const float* __restrict__ arow = A + (m * Kd + half * 2);
  const float* __restrict__ bcol = B + (half * 2 * NP + n);

  v8f acc = {};
#pragma unroll
  for (int k0 = 0; k0 < Kd; k0 += 4) {
    v2f a = *(const v2f*)(arow + k0);
    v2f b;
    b[0] = bcol[(k0 + 0) * NP];
    b[1] = bcol[(k0 + 1) * NP];
    acc = wmma_f32(a, b, acc);
  }

#pragma unroll
  for (int r = 0; r < 8; ++r) {
    const int mm = mt * 16 + r + half * 8;
    float v = acc[r];
    if constexpr (GELU) v = gelu_tanh(v);
    if constexpr (RES) v += resid[mm * NP + n];
    outp[mm * NP + n] = v;
  }
}

// ---------------------------------------------------------------------------
// Bilinear 2x upsample, half-pixel centers, edge clamp.
// ---------------------------------------------------------------------------
__global__ void upsample_bilinear_kernel(const float* __restrict__ in,
                                         float* __restrict__ outp) {
  constexpr int C = 32, Hin = 128, Win = 256, Hout = 256, Wout = 512;
  int idx = blockIdx.x * blockDim.x + threadIdx.x;
  if (idx >= C * Hout * Wout) return;
  int wo = idx & (Wout - 1);
  int t = idx >> 9;  // / Wout
  int ho = t & (Hout - 1);
  int c = t >> 8;  // / Hout

  float fi = (ho + 0.5f) * 0.5f - 0.5f;
  float fj = (wo + 0.5f) * 0.5f - 0.5f;
  float fi0 = floorf(fi), fj0 = floorf(fj);
  float di = fi - fi0, dj = fj - fj0;
  int i0 = (int)fi0, j0 = (int)fj0;
  int i1 = i0 + 1, j1 = j0 + 1;
  i0 = i0 < 0 ? 0 : (i0 > Hin - 1 ? Hin - 1 : i0);
  i1 = i1 < 0 ? 0 : (i1 > Hin - 1 ? Hin - 1 : i1);
  j0 = j0 < 0 ? 0 : (j0 > Win - 1 ? Win - 1 : j0);
  j1 = j1 < 0 ? 0 : (j1 > Win - 1 ? Win - 1 : j1);

  const float* p = in + c * Hin * Win;
  float v00 = p[i0 * Win + j0];
  float v01 = p[i0 * Win + j1];
  float v10 = p[i1 * Win + j0];
  float v11 = p[i1 * Win + j1];
  float v0 = v00 + dj * (v01 - v00);
  float v1 = v10 + dj * (v11 - v10);
  outp[idx] = v0 + di * (v1 - v0);
}

// ---------------------------------------------------------------------------
// Host-side orchestration.
// ---------------------------------------------------------------------------
extern "C" void kernel_launch(void* const* d_in, const int* in_sizes, int n_in,
                              void* d_out, int out_size, void* d_ws, size_t ws_size,
                              hipStream_t stream) {
  (void)in_sizes; (void)n_in; (void)out_size; (void)ws_size;

  const float* x      = (const float*)d_in[0];   // [1,3,256,512]
  const float* pe_v   = (const float*)d_in[1];   // [9,128,16]
  const int*   pe_lat = (const int*)  d_in[2];   // [128,16]
  const int*   pe_off = (const int*)  d_in[3];   // [128,16]
  const float* w_enc  = (const float*)d_in[4];   // [32,3,9]
  const float* pb_v   = (const float*)d_in[5];   // [9,128,16]
  const int*   pb_lat = (const int*)  d_in[6];   // [128,16]
  const int*   pb_off = (const int*)  d_in[7];   // [128,16]
  const float* w_blk  = (const float*)d_in[8];   // [4,32,32,9]
  const float* mlp_w1 = (const float*)d_in[9];   // [4,64,32]
  const float* mlp_w2 = (const float*)d_in[10];  // [4,32,64]
  const float* pd_v   = (const float*)d_in[11];  // [9,256,16]
  const int*   pd_lat = (const int*)  d_in[12];  // [256,16]
  const int*   pd_off = (const int*)  d_in[13];  // [256,16]
  const float* w_dec  = (const float*)d_in[14];  // [3,32,9]

  const int NP = 128 * 256;

  float* ws  = (float*)d_ws;
  float* zA  = ws;                 // 1,048,576 floats
  float* zB  = ws + 1048576;       // 1,048,576
  float* yb  = ws + 2097152;       // 1,048,576
  float* hid = ws + 3145728;       // 2,097,152
  float* wv  = ws + 5242880;       // 2,097,152 (reused enc/blk/dec; dec padded O=16 fits)
  float* zu  = zB;                 // 4,194,304: aliases dead zB+yb+hid after layers

  // ---- encoder: [3,256,512] -> z [32,128,256], stride 2 ----
  {
    int total = 128 * 32 * (3 * 16);
    build_wv_kernel<<<(total + 255) / 256, 256, 0, stream>>>(w_enc, pe_v, wv, 32, 32, 3, 128);
    disco_wmma_conv_kernel<32, 3, 256, 512, 128, 256, 2>
        <<<dim3(16, 2, 128), 32, 0, stream>>>(x, wv, pe_lat, pe_off, zA);
  }

  // ---- 4 layers ----
  float* zin = zA;
  float* zout = zB;
  for (int i = 0; i < 4; ++i) {
    int total = 128 * 32 * (32 * 16);
    build_wv_kernel<<<(total + 255) / 256, 256, 0, stream>>>(
        w_blk + (size_t)i * 32 * 32 * 9, pb_v, wv, 32, 32, 32, 128);
    disco_wmma_conv_kernel<32, 32, 128, 256, 128, 256, 1>
        <<<dim3(16, 2, 128), 32, 0, stream>>>(zin, wv, pb_lat, pb_off, yb);
    // hidden = gelu(w1 @ y)
    mlp_gemm_kernel<64, 32, true, false><<<dim3(NP / 16, 4), 32, 0, stream>>>(
        mlp_w1 + (size_t)i * 64 * 32, yb, hid, nullptr);
    // z_out = w2 @ hidden + z_in
    mlp_gemm_kernel<32, 64, false, true><<<dim3(NP / 16, 2), 32, 0, stream>>>(
        mlp_w2 + (size_t)i * 32 * 64, hid, zout, zin);
    float* t = zin; zin = zout; zout = t;
  }
  // after 4 swaps, final z is in zA (== zin)

  // ---- upsample 2x: [32,128,256] -> [32,256,512] ----
  {
    int total = 32 * 256 * 512;
    upsample_bilinear_kernel<<<(total + 255) / 256, 256, 0, stream>>>(zin, zu);
  }

  // ---- decoder: [32,256,512] -> out [3,256,512] ----
  {
    int total = 256 * 16 * (32 * 16);  // padded O = 16
    build_wv_kernel<<<(total + 255) / 256, 256, 0, stream>>>(w_dec, pd_v, wv, 3, 16, 32, 256);
    disco_wmma_conv_kernel<3, 32, 256, 512, 256, 512, 1>
        <<<dim3(32, 1, 256), 32, 0, stream>>>(zu, wv, pd_lat, pd_off, (float*)d_out);
  }
}